// RippleNet_12781822673189
// MI455X (gfx1250) — compile-verified
//
#include <hip/hip_runtime.h>

#define N_NODES 16384
#define N_MEM   32
#define DIM     16

typedef __attribute__((ext_vector_type(2))) float v2f;
typedef __attribute__((ext_vector_type(8))) float v8f;

// ---------------------------------------------------------------------------
// Kernel 1: item[n, :] = entity_emb[nodes[n], :]
// ---------------------------------------------------------------------------
__global__ __launch_bounds__(256)
void ripple_init_item(const int* __restrict__ nodes,
                      const float* __restrict__ entity_emb,
                      float* __restrict__ item) {
    int idx = blockIdx.x * blockDim.x + threadIdx.x;   // over N_NODES*DIM
    if (idx >= N_NODES * DIM) return;
    int n = idx >> 4;
    int d = idx & 15;
    item[idx] = entity_emb[nodes[n] * DIM + d];
}

// ---------------------------------------------------------------------------
// Kernel 2: one wave32 per node; lane = memory slot.
//   score_m = item . (R_m @ h_m)  (256 fused MACs per lane, fully unrolled)
//   softmax over 32 lanes via xor-butterfly shuffles
//   o = sum_m p_m * t_m           (16-float vector butterfly reduce)
//   hop 1 fuses the final combine: out = o1 + prev_scale * o0
// ---------------------------------------------------------------------------
__global__ __launch_bounds__(32)
void ripple_hop(const int* __restrict__ mh, const int* __restrict__ mr,
                const int* __restrict__ mt,
                const float* __restrict__ entity_emb,
                const float* __restrict__ relation_emb,
                const float* __restrict__ item,
                const float* __restrict__ o_prev, float prev_scale,
                float* __restrict__ o_out) {
    const int n    = blockIdx.x;
    const int lane = threadIdx.x;            // memory slot m (wave32)
    const int base = n * N_MEM + lane;

    const int hidx = mh[base];
    const int ridx = mr[base];
    const int tidx = mt[base];

    // item vector for this node (broadcast load, L2/WGP$ resident)
    const float* __restrict__ ip = item + n * DIM;
    float it[DIM];
#pragma unroll
    for (int d = 0; d < DIM; ++d) it[d] = ip[d];

    // head embedding gather
    const float* __restrict__ hp = entity_emb + (long)hidx * DIM;
    float h[DIM];
#pragma unroll
    for (int d = 0; d < DIM; ++d) h[d] = hp[d];

    // relation matrix gather: 64KB table -> cache hits; prefetch the row
    const float* __restrict__ Rp = relation_emb + (long)ridx * (DIM * DIM);
    __builtin_prefetch(Rp, 0, 0);

    // score = sum_d it[d] * (R[d,:] . h)
    float score = 0.f;
#pragma unroll
    for (int d = 0; d < DIM; ++d) {
        float acc = 0.f;
#pragma unroll
        for (int e = 0; e < DIM; ++e)
            acc = __builtin_fmaf(Rp[d * DIM + e], h[e], acc);
        score = __builtin_fmaf(it[d], acc, score);
    }

    // ---- softmax over the 32 memory slots (wave32 butterfly) ----
    float mx = score;
#pragma unroll
    for (int off = 16; off > 0; off >>= 1)
        mx = fmaxf(mx, __shfl_xor(mx, off, 32));
    float ex = __expf(score - mx);
    float sum = ex;
#pragma unroll
    for (int off = 16; off > 0; off >>= 1)
        sum += __shfl_xor(sum, off, 32);
    const float p = ex / sum;

    // ---- weighted tail sum: o[d] = sum_m p_m * t_m[d] ----
    const float* __restrict__ tp = entity_emb + (long)tidx * DIM;
    float o[DIM];
#pragma unroll
    for (int d = 0; d < DIM; ++d) o[d] = p * tp[d];
#pragma unroll
    for (int off = 16; off > 0; off >>= 1) {
#pragma unroll
        for (int d = 0; d < DIM; ++d)
            o[d] += __shfl_xor(o[d], off, 32);
    }

    if (lane == 0) {
        float* __restrict__ op = o_out + n * DIM;
        if (o_prev) {
            const float* __restrict__ pp = o_prev + n * DIM;
#pragma unroll
            for (int d = 0; d < DIM; ++d)
                op[d] = __builtin_fmaf(prev_scale, pp[d], o[d]);
        } else {
#pragma unroll
            for (int d = 0; d < DIM; ++d) op[d] = o[d];
        }
    }
}

// ---------------------------------------------------------------------------
// Kernel 3: item = (item + o) @ W^T via V_WMMA_F32_16X16X4_F32.
// One wave32 per 16-node tile; K=16 done as 4 chained K=4 WMMAs.
//   D[m=node, n=dim] = sum_k A[m,k] * B[k,n],  A = item+o,  B[k,n] = W[n,k].
// Fragment layouts per CDNA5 ISA 7.12.2:
//   A 16x4 : lanes 0-15 -> M=lane, VGPR{0,1}=K{0,1}; lanes 16-31 -> K{2,3}
//   B 4x16 : lanes 0-15 -> N=lane, VGPR{0,1}=K{0,1}; lanes 16-31 -> K{2,3}
//   C/D    : lane col = lane&15, VGPR j = row j + (lane>=16 ? 8 : 0)
// ---------------------------------------------------------------------------
__global__ __launch_bounds__(32)
void ripple_item_update(const float* __restrict__ o,
                        const float* __restrict__ W,
                        float* __restrict__ item) {
    const int lane  = threadIdx.x;
    const int rc    = lane & 15;      // A-row (node) / B,C,D-column (dim)
    const int hi    = lane >> 4;      // 0 or 1: selects K/M half
    const int node0 = blockIdx.x * 16;

    const float* __restrict__ Si = item + node0 * DIM;
    const float* __restrict__ So = o    + node0 * DIM;

    v8f c = {};
#pragma unroll
    for (int kc = 0; kc < 4; ++kc) {
        const int kb = kc * 4 + hi * 2;
        v2f a, b;
        a.x = Si[rc * DIM + kb]     + So[rc * DIM + kb];
        a.y = Si[rc * DIM + kb + 1] + So[rc * DIM + kb + 1];
        b.x = W[rc * DIM + kb];       // B[k][n] = W[n][k], n = rc
        b.y = W[rc * DIM + kb + 1];
        // 8 args: (neg_a, A, neg_b, B, c_mod, C, reuse_a, reuse_b)
        c = __builtin_amdgcn_wmma_f32_16x16x4_f32(false, a, false, b,
                                                  (short)0, c, false, false);
    }

    // write back D: in-place update of the same 16 rows this block read
    float* __restrict__ out = item + node0 * DIM;
#pragma unroll
    for (int j = 0; j < 8; ++j)
        out[(j + hi * 8) * DIM + rc] = c[j];
}

// ---------------------------------------------------------------------------
// Host-side launcher. Inputs (setup_inputs order):
//  0 nodes[i32 16384], 1 memories_h[i32 2*16384*32], 2 memories_r, 3 memories_t,
//  4 entity_emb[f32 100000*16], 5 relation_emb[f32 64*256], 6 W[f32 16*16]
// Output: f32 [16384*16].  Workspace: item (1MB) + o0 (1MB).
// ---------------------------------------------------------------------------
extern "C" void kernel_launch(void* const* d_in, const int* in_sizes, int n_in,
                              void* d_out, int out_size, void* d_ws, size_t ws_size,
                              hipStream_t stream) {
    const int*   nodes = (const int*)d_in[0];
    const int*   mh    = (const int*)d_in[1];
    const int*   mr    = (const int*)d_in[2];
    const int*   mt    = (const int*)d_in[3];
    const float* ent   = (const float*)d_in[4];
    const float* rel   = (const float*)d_in[5];
    const float* W     = (const float*)d_in[6];
    float* out = (float*)d_out;

    float* item = (float*)d_ws;               // N_NODES*DIM floats
    float* o0   = item + N_NODES * DIM;       // N_NODES*DIM floats

    // item = entity_emb[nodes]
    ripple_init_item<<<(N_NODES * DIM + 255) / 256, 256, 0, stream>>>(nodes, ent, item);

    // hop 0 -> o0
    ripple_hop<<<N_NODES, 32, 0, stream>>>(mh, mr, mt, ent, rel, item,
                                           nullptr, 0.0f, o0);

    // item = (item + o0) @ W^T   (WMMA)
    ripple_item_update<<<N_NODES / 16, 32, 0, stream>>>(o0, W, item);

    // hop 1 -> out = o1 + 2*o0   (fused final combine)
    const int off = N_NODES * N_MEM;
    ripple_hop<<<N_NODES, 32, 0, stream>>>(mh + off, mr + off, mt + off, ent, rel, item,
                                           o0, 2.0f, out);
}